// WithLSHSort_22308060135932
// MI455X (gfx1250) — compile-verified
//
#include <hip/hip_runtime.h>
#include <cstdint>
#include <cstddef>

// WithLSHSort reduces to the identity: idx = argsort(angles), inv = argsort(idx)
// = idx^-1, and the two take_along_axis gathers cancel (y == x_sorted with no
// transform in between). out == x bitwise; W/b/angles are dead code.
// Optimal MI455X kernel = HBM-rate copy of 64 MiB in + 64 MiB out (~5.5 us at
// 23.3 TB/s; both buffers fit the 192 MB L2, so graph replays may run faster).
//
// Copy rides the CDNA5 async Global->LDS->Global DMA path (ASYNCcnt-tracked,
// b128 per lane). Per the ISA, INST_OFFSET is added to BOTH the LDS and global
// addresses, so the per-k chunk offset is folded into the offset: immediate --
// one base address per direction, zero per-op address arithmetic.

#define COPY_THREADS 256
#define COPY_UNROLL  8
#define CHUNK_BYTES  (COPY_THREADS * COPY_UNROLL * 16)  // 32 KB per block

__global__ __launch_bounds__(COPY_THREADS)
void lsh_identity_async_copy(const float* __restrict__ x,
                             float* __restrict__ out)
{
    __shared__ __align__(16) char lds_buf[CHUNK_BYTES];

    const uint32_t tid        = threadIdx.x;
    const size_t   chunk_base = (size_t)blockIdx.x * (size_t)CHUNK_BYTES;

    // Per-thread bases; each async op below adds k*4096 via offset: immediate,
    // which applies to both the LDS and the global side in lockstep.
    const void* src_base = (const void*)((const char*)x + chunk_base + (size_t)tid * 16u);
    void*       dst_base = (void*)((char*)out + chunk_base + (size_t)tid * 16u);

    // Low 32 bits of the generic address of an LDS object == LDS byte offset
    // (aperture scheme).
    const uint32_t lds_base = (uint32_t)(uintptr_t)(void*)&lds_buf[0] + tid * 16u;

    // Stage 1: async DMA global -> LDS, 128b per lane, 8 in flight per thread.
#pragma unroll
    for (int k = 0; k < COPY_UNROLL; ++k) {
        asm volatile("global_load_async_to_lds_b128 %0, %1, off offset:%c2"
                     :
                     : "v"(lds_base), "v"(src_base), "i"(k * COPY_THREADS * 16)
                     : "memory");
    }

    // All of this wave's async loads have landed in LDS.
    asm volatile("s_wait_asynccnt 0" ::: "memory");

    // Stage 2: async DMA LDS -> global. Drained by implicit wait at s_endpgm.
#pragma unroll
    for (int k = 0; k < COPY_UNROLL; ++k) {
        asm volatile("global_store_async_from_lds_b128 %0, %1, off offset:%c2"
                     :
                     : "v"(dst_base), "v"(lds_base), "i"(k * COPY_THREADS * 16)
                     : "memory");
    }
}

// Fallback for shapes that don't tile into 32 KB chunks (not hit for the
// reference shape, but keeps kernel_launch total & deterministic).
__global__ void lsh_identity_copy_f32(const float* __restrict__ x,
                                      float* __restrict__ out, size_t n)
{
    size_t i      = (size_t)blockIdx.x * blockDim.x + threadIdx.x;
    size_t stride = (size_t)gridDim.x * blockDim.x;
    for (; i < n; i += stride) out[i] = x[i];
}

extern "C" void kernel_launch(void* const* d_in, const int* in_sizes, int n_in,
                              void* d_out, int out_size, void* d_ws, size_t ws_size,
                              hipStream_t stream)
{
    (void)n_in; (void)out_size; (void)d_ws; (void)ws_size;

    const float* x   = (const float*)d_in[0];   // (B,S,D) fp32, B*S*D elements
    float*       out = (float*)d_out;           // same size, fp32

    const size_t n     = (size_t)in_sizes[0];   // 4*4096*1024 = 16,777,216
    const size_t bytes = n * sizeof(float);     // 64 MiB

    const bool aligned = ((((uintptr_t)x) | ((uintptr_t)out)) & 0xF) == 0;

    if (aligned && (bytes % CHUNK_BYTES) == 0) {
        const int blocks = (int)(bytes / CHUNK_BYTES);  // 2048 for ref shape
        lsh_identity_async_copy<<<blocks, COPY_THREADS, 0, stream>>>(x, out);
    } else {
        const int blocks = 2048;
        lsh_identity_copy_f32<<<blocks, 256, 0, stream>>>(x, out, n);
    }
}